// GNNBasicBlock_31121333027067
// MI455X (gfx1250) — compile-verified
//
#include <hip/hip_runtime.h>
#include <math.h>

typedef __attribute__((ext_vector_type(2))) float v2f;
typedef __attribute__((ext_vector_type(8))) float v8f;

#define FD 128  // feature dim, fixed by the reference

// ---------------------------------------------------------------- utilities
__global__ void zero_kernel(float* __restrict__ p, size_t n) {
    size_t i = (size_t)blockIdx.x * blockDim.x + threadIdx.x;
    size_t stride = (size_t)gridDim.x * blockDim.x;
    for (; i < n; i += stride) p[i] = 0.0f;
}

// one thread per edge: accumulate out/in degrees with native f32 atomics
__global__ void degree_kernel(const int* __restrict__ src, const int* __restrict__ dst,
                              float* __restrict__ degS, float* __restrict__ degD, int E) {
    int e = blockIdx.x * blockDim.x + threadIdx.x;
    if (e >= E) return;
    __hip_atomic_fetch_add(&degS[src[e]], 1.0f, __ATOMIC_RELAXED, __HIP_MEMORY_SCOPE_AGENT);
    __hip_atomic_fetch_add(&degD[dst[e]], 1.0f, __ATOMIC_RELAXED, __HIP_MEMORY_SCOPE_AGENT);
}

// deg -> deg^{-1/2} with clamp-to-1, in place (covers both arrays back to back)
__global__ void norm_kernel(float* __restrict__ p, int n) {
    int i = blockIdx.x * blockDim.x + threadIdx.x;
    if (i < n) p[i] = rsqrtf(fmaxf(p[i], 1.0f));
}

// one wave32 per edge: agg[dst] += x[src] * norm_src[src]  (float4 per lane)
__global__ __launch_bounds__(256) void scatter_kernel(
    const float* __restrict__ x, const int* __restrict__ src, const int* __restrict__ dst,
    const float* __restrict__ normS, float* __restrict__ agg, int E) {
    int e = blockIdx.x * 8 + (threadIdx.x >> 5);
    if (e >= E) return;
    const int lane = threadIdx.x & 31;
    const int s = src[e];
    const int d = dst[e];
    const float ns = normS[s];
    const float4 xv = *(const float4*)(x + (size_t)s * FD + lane * 4);
    float* ap = agg + (size_t)d * FD + lane * 4;
    __hip_atomic_fetch_add(ap + 0, xv.x * ns, __ATOMIC_RELAXED, __HIP_MEMORY_SCOPE_AGENT);
    __hip_atomic_fetch_add(ap + 1, xv.y * ns, __ATOMIC_RELAXED, __HIP_MEMORY_SCOPE_AGENT);
    __hip_atomic_fetch_add(ap + 2, xv.z * ns, __ATOMIC_RELAXED, __HIP_MEMORY_SCOPE_AGENT);
    __hip_atomic_fetch_add(ap + 3, xv.w * ns, __ATOMIC_RELAXED, __HIP_MEMORY_SCOPE_AGENT);
}

// ---------------------------------------------------------------- fused GEMM + NodeNorm
// block = 256 threads = 8 waves; block owns a 16-node row tile, wave w owns cols [16w,16w+16)
// h = (agg * norm_dst) @ W + bias ; nodenorm ; relu ; + x
__global__ __launch_bounds__(256) void gemm_nodenorm_kernel(
    const float* __restrict__ agg, const float* __restrict__ normD,
    const float* __restrict__ W, const float* __restrict__ bias,
    const float* __restrict__ x, float* __restrict__ out, int N) {
    __shared__ float ldsA[16 * FD];
    __shared__ float ldsH[16 * FD];

    const int rowBase = blockIdx.x * 16;

    // stage A tile (scaled by norm_dst) into LDS, coalesced float4
    for (int t = threadIdx.x; t < (16 * FD) / 4; t += 256) {
        const int idx = t * 4;
        const int m = idx >> 7;           // idx / 128
        int row = rowBase + m;
        if (row > N - 1) row = N - 1;     // clamp (N % 16 == 0 for this problem)
        const float nd = normD[row];
        const float4 v = *(const float4*)(agg + (size_t)row * FD + (idx & (FD - 1)));
        ldsA[idx + 0] = v.x * nd;
        ldsA[idx + 1] = v.y * nd;
        ldsA[idx + 2] = v.z * nd;
        ldsA[idx + 3] = v.w * nd;
    }
    __syncthreads();

    const int lane = threadIdx.x & 31;
    const int wave = threadIdx.x >> 5;           // 0..7 -> column tile
    const int nCol = wave * 16 + (lane & 15);    // global output column (D == 128)
    const int mRow = lane & 15;                  // A-matrix row within tile
    const int kOff = (lane >> 4) * 2;            // lanes 16-31 hold K = k0+2, k0+3

    v8f acc;
    {
        const float b0 = bias[nCol];             // same column for all 8 acc rows
#pragma unroll
        for (int i = 0; i < 8; ++i) acc[i] = b0;
    }

#pragma unroll 4
    for (int k0 = 0; k0 < FD; k0 += 4) {
        const int k = k0 + kOff;
        const v2f a = *(const v2f*)&ldsA[mRow * FD + k];   // ds_load_b64, 8B aligned
        v2f b;
        b.x = W[(size_t)k * FD + nCol];                    // coalesced 64B across lanes
        b.y = W[(size_t)(k + 1) * FD + nCol];
        acc = __builtin_amdgcn_wmma_f32_16x16x4_f32(false, a, false, b, (short)0, acc,
                                                    false, false);
    }

    // spill h tile to LDS: acc[i] is row (i + 8*(lane>=16)), column nCol
    {
        const int mHi = (lane >> 4) * 8;
#pragma unroll
        for (int i = 0; i < 8; ++i) ldsH[(mHi + i) * FD + nCol] = acc[i];
    }
    __syncthreads();

    // NodeNorm + relu + residual: 16 threads per row, 8 cols each
    const int row = threadIdx.x >> 4;
    const int l16 = threadIdx.x & 15;
    float vals[8];
    float s = 0.f, s2 = 0.f;
#pragma unroll
    for (int j = 0; j < 8; ++j) {
        const float v = ldsH[row * FD + l16 + j * 16];
        vals[j] = v;
        s += v;
        s2 += v * v;
    }
#pragma unroll
    for (int m = 8; m >= 1; m >>= 1) {           // reduce within 16-lane group (wave32)
        s += __shfl_xor(s, m, 16);
        s2 += __shfl_xor(s2, m, 16);
    }
    const float mean = s * (1.0f / FD);
    const float var = s2 * (1.0f / FD) - mean * mean;
    const float inv = rsqrtf(var + 1e-5f);

    const int node = rowBase + row;
    if (node < N) {
        const size_t base = (size_t)node * FD;
#pragma unroll
        for (int j = 0; j < 8; ++j) {
            const int c = l16 + j * 16;
            float v = (vals[j] - mean) * inv;
            v = v > 0.0f ? v : 0.0f;
            out[base + c] = v + x[base + c];
        }
    }
}

// ---------------------------------------------------------------- launch
extern "C" void kernel_launch(void* const* d_in, const int* in_sizes, int n_in,
                              void* d_out, int out_size, void* d_ws, size_t ws_size,
                              hipStream_t stream) {
    (void)n_in; (void)out_size; (void)ws_size;
    const float* x    = (const float*)d_in[0];
    const float* W    = (const float*)d_in[1];
    const float* bias = (const float*)d_in[2];
    const int*   src  = (const int*)d_in[3];
    const int*   dst  = (const int*)d_in[4];
    const int N = in_sizes[0] / FD;
    const int E = in_sizes[3];

    float* agg   = (float*)d_ws;                    // N * 128
    float* normS = agg + (size_t)N * FD;            // N   (out-degree -> norm_src)
    float* normD = normS + N;                       // N   (in-degree  -> norm_dst)
    float* out   = (float*)d_out;

    const size_t zn = (size_t)N * FD + 2 * (size_t)N;
    zero_kernel<<<2048, 256, 0, stream>>>(agg, zn);
    degree_kernel<<<(E + 255) / 256, 256, 0, stream>>>(src, dst, normS, normD, E);
    norm_kernel<<<(2 * N + 255) / 256, 256, 0, stream>>>(normS, 2 * N);
    scatter_kernel<<<(E + 7) / 8, 256, 0, stream>>>(x, src, dst, normS, agg, E);
    gemm_nodenorm_kernel<<<(N + 15) / 16, 256, 0, stream>>>(agg, normD, W, bias, x, out, N);
}